// Router_67860483276966
// MI455X (gfx1250) — compile-verified
//
#include <hip/hip_runtime.h>

// CDNA5 / gfx1250: wave32, WMMA f32 16x16x4 path (fp32 data, HBM-bound kernel).

typedef float v2f __attribute__((ext_vector_type(2)));
typedef float v8f __attribute__((ext_vector_type(8)));

#define NREG     4096
#define DIM      64
#define DEG      6
#define MFREQ    8
#define FB_ALPHA 0.1f
#define FB_SCALE 0.35355339059327378f   // 1/sqrt(8)

// One wave (32 lanes) per destination node r.
//   y_r = sum_{e in [6r,6r+6)} (W_e @ (w_e * H[src_e]))
// 64x64 matvec -> 4 row-tiles x 16 k-tiles of V_WMMA_F32_16X16X4_F32,
// with the h-vector broadcast across all 16 N columns of B.
__global__ __launch_bounds__(256) void router_wmma_f32(
    const float* __restrict__ H,            // [R, 64]
    const unsigned char* __restrict__ mask, // [R] (bool)
    const float* __restrict__ coords,       // [R, 2]
    const float* __restrict__ Wedges,       // [E, 64, 64] row-major [out,in]
    const float* __restrict__ Wreg,         // [8, 2]
    const float* __restrict__ bcos,         // [8]
    const float* __restrict__ bsin,         // [8]
    const int* __restrict__ src,            // [E]
    float* __restrict__ out)                // [R, 64]
{
    const int lane = threadIdx.x & 31;
    const int wave = threadIdx.x >> 5;
    const int r    = blockIdx.x * 8 + wave;     // node id (grid sized exactly)
    const int half = lane >> 4;                 // 0: lanes 0-15, 1: lanes 16-31
    const int m16  = lane & 15;                 // M index within a 16-row tile

    v8f c[4];
    #pragma unroll
    for (int it = 0; it < 4; ++it) c[it] = (v8f){0,0,0,0,0,0,0,0};

    const float prx = coords[2 * r];
    const float pry = coords[2 * r + 1];

    #pragma unroll 1
    for (int e = 0; e < DEG; ++e) {
        const int eidx = r * DEG + e;
        const int s    = src[eidx];

        // Prefetch next edge's 16KB weight block while this one computes.
        if (e + 1 < DEG) {
            const char* wn = (const char*)(Wedges + (size_t)(eidx + 1) * (DIM * DIM));
            __builtin_prefetch(wn + (size_t)lane * 512, 0, 0);   // 32 lanes x 512B = 16KB
        }

        // ---- per-edge scalar weight w_e (computed redundantly per lane) ----
        const float dx = prx - coords[2 * s];
        const float dy = pry - coords[2 * s + 1];
        float acc = 0.0f;
        #pragma unroll
        for (int m = 0; m < MFREQ; ++m) {
            const float sv = dx * Wreg[2 * m] + dy * Wreg[2 * m + 1];
            acc += __cosf(sv) * bcos[m] + __sinf(sv) * bsin[m];
        }
        const float w = mask[s] ? (1.0f + FB_ALPHA * FB_SCALE * acc) : 0.0f;

        // ---- B operands: scaled h broadcast across N, one v2f per k-tile ----
        // B 4x16 layout: vgpr0 = row K=0 (lanes 0-15) / K=2 (lanes 16-31),
        //                vgpr1 = row K=1 / K=3  -> float2 at h[k0 + 2*half].
        v2f b[16];
        const float* hs = H + (size_t)s * DIM;
        #pragma unroll
        for (int kt = 0; kt < 16; ++kt) {
            v2f hv = *(const v2f*)(hs + kt * 4 + 2 * half);
            b[kt].x = hv.x * w;
            b[kt].y = hv.y * w;
        }

        // ---- A tiles + WMMA accumulate ----
        // A 16x4 layout: lane M = lane%16; vgpr0/1 hold K = 2*half, 2*half+1.
        const float* wbase = Wedges + (size_t)eidx * (DIM * DIM)
                           + (size_t)m16 * DIM + 2 * half;
        #pragma unroll
        for (int kt = 0; kt < 16; ++kt) {
            #pragma unroll
            for (int it = 0; it < 4; ++it) {
                v2f a = __builtin_nontemporal_load(
                            (const v2f*)(wbase + (size_t)it * 16 * DIM + kt * 4));
                c[it] = __builtin_amdgcn_wmma_f32_16x16x4_f32(
                            false, a, false, b[kt], (short)0, c[it], false, false);
            }
        }
    }

    // D layout: vgpr v, lanes 0-15 -> M = v, lanes 16-31 -> M = v+8; all N
    // columns identical, so lanes 0 and 16 each own 8 consecutive rows.
    if (m16 == 0) {
        float* op = out + (size_t)r * DIM + half * 8;
        #pragma unroll
        for (int it = 0; it < 4; ++it) {
            *(v8f*)(op + it * 16) = c[it];
        }
    }
}

extern "C" void kernel_launch(void* const* d_in, const int* in_sizes, int n_in,
                              void* d_out, int out_size, void* d_ws, size_t ws_size,
                              hipStream_t stream) {
    const float*         H      = (const float*)d_in[0];
    const unsigned char* mask   = (const unsigned char*)d_in[1];  // bool
    const float*         coords = (const float*)d_in[2];
    const float*         Wedges = (const float*)d_in[3];
    const float*         Wreg   = (const float*)d_in[4];
    const float*         bcos   = (const float*)d_in[5];
    const float*         bsin   = (const float*)d_in[6];
    const int*           src    = (const int*)d_in[7];
    // d_in[8] = dst: unused, dst[e] == e / DEG by construction (sorted).

    // 8 waves per block, one wave per node: 4096 / 8 = 512 blocks.
    router_wmma_f32<<<NREG / 8, 256, 0, stream>>>(
        H, mask, coords, Wedges, Wreg, bcos, bsin, src, (float*)d_out);
}